// YOLOF_58360015618085
// MI455X (gfx1250) — compile-verified
//
#include <hip/hip_runtime.h>
#include <stdint.h>

#define NUM_CLASSES 80
#define TOPK        1000
#define NBINS       4096
#define CAND_MAX    4096
#define STRIDE_F    32.0f
#define CONF_THRESH 0.05f
#define NMS_THRESH  0.6f
#define CTR_CLAMP   32.0f
#define SCALE_CLAMP 4.13516655674236f   /* log(1000/16) */
#define CLASS_OFFSET 100000.0f

typedef float v2f __attribute__((ext_vector_type(2)));
typedef float v8f __attribute__((ext_vector_type(8)));

__device__ __forceinline__ float sigmoidf_(float x) {
    return 1.0f / (1.0f + expf(-x));
}

// ---------------- Stage 0: zero histogram + counter ----------------
__global__ void kZero(int* hist, int* counter) {
    int idx = blockIdx.x * blockDim.x + threadIdx.x;
    if (idx < NBINS) hist[idx] = 0;
    else if (idx == NBINS) *counter = 0;
}

// ---------------- Stage 1: score histogram ----------------
__global__ void kHist(const float* __restrict__ cls, int n, int* __restrict__ hist) {
    int idx = blockIdx.x * blockDim.x + threadIdx.x;
    if (idx >= n) return;
    float s = sigmoidf_(cls[idx]);
    int b = (int)(s * (float)NBINS);
    b = b < 0 ? 0 : (b > NBINS - 1 ? NBINS - 1 : b);
    atomicAdd(&hist[b], 1);
}

// ---------------- Stage 2: find cutoff bin (serial, tiny) ----------------
__global__ void kCutoff(const int* __restrict__ hist, int* __restrict__ cutoff) {
    if (threadIdx.x != 0 || blockIdx.x != 0) return;
    int total = 0, cb = 0;
    for (int b = NBINS - 1; b >= 0; --b) {
        total += hist[b];
        if (total >= TOPK) { cb = b; break; }
    }
    *cutoff = cb;
}

// ---------------- Stage 3: collect candidates ----------------
__global__ void kCollect(const float* __restrict__ cls, int n,
                         const int* __restrict__ cutoff,
                         unsigned long long* __restrict__ cand,
                         int* __restrict__ counter) {
    int idx = blockIdx.x * blockDim.x + threadIdx.x;
    if (idx >= n) return;
    float s = sigmoidf_(cls[idx]);
    int b = (int)(s * (float)NBINS);
    b = b < 0 ? 0 : (b > NBINS - 1 ? NBINS - 1 : b);
    if (b >= *cutoff) {
        unsigned long long key =
            ((unsigned long long)__float_as_uint(s) << 32) |
            (unsigned long long)(0xFFFFFFFFu - (unsigned)idx);
        int pos = atomicAdd(counter, 1);
        if (pos < CAND_MAX) cand[pos] = key;
    }
}

// ---------------- Stage 4: bitonic sort in LDS + decode top-1000 ----------------
__global__ __launch_bounds__(1024) void kSortDecode(
    const unsigned long long* __restrict__ cand, const int* __restrict__ counter,
    const float* __restrict__ reg_pred, const float* __restrict__ anchor_size,
    const int* __restrict__ fmp_w_p, int ka,
    float* __restrict__ prob_ws, int* __restrict__ label_ws,
    float* __restrict__ boxes_ws, float* __restrict__ boxoff_ws,
    float* __restrict__ areas_ws)
{
    __shared__ unsigned long long sk[CAND_MAX];
    const int tid = threadIdx.x;
    int count = *counter;
    if (count > CAND_MAX) count = CAND_MAX;
    for (int i = tid; i < CAND_MAX; i += 1024) sk[i] = (i < count) ? cand[i] : 0ull;
    __syncthreads();

    // bitonic sort, descending (key = score_bits<<32 | ~idx  => score desc, idx asc)
    for (unsigned k = 2; k <= CAND_MAX; k <<= 1) {
        for (unsigned j = k >> 1; j > 0; j >>= 1) {
            for (unsigned idx = tid; idx < CAND_MAX; idx += 1024) {
                unsigned ixj = idx ^ j;
                if (ixj > idx) {
                    unsigned long long a = sk[idx], b = sk[ixj];
                    bool up = ((idx & k) == 0);
                    if (up ? (a < b) : (a > b)) { sk[idx] = b; sk[ixj] = a; }
                }
            }
            __syncthreads();
        }
    }

    if (tid < TOPK) {
        unsigned long long key = sk[tid];
        float s = __uint_as_float((unsigned)(key >> 32));
        unsigned flat = 0xFFFFFFFFu - (unsigned)(key & 0xFFFFFFFFull);
        int aidx = (int)(flat / NUM_CLASSES);
        int label = (int)(flat % NUM_CLASSES);
        int p  = aidx / ka;
        int kk = aidx - p * ka;
        int fw = *fmp_w_p;
        int gx = p % fw, gy = p / fw;
        float cx = ((float)gx + 0.5f) * STRIDE_F;
        float cy = ((float)gy + 0.5f) * STRIDE_F;
        float aw = anchor_size[kk * 2 + 0];
        float ah = anchor_size[kk * 2 + 1];
        float r0 = reg_pred[aidx * 4 + 0], r1 = reg_pred[aidx * 4 + 1];
        float r2 = reg_pred[aidx * 4 + 2], r3 = reg_pred[aidx * 4 + 3];
        float ox = fminf(fmaxf(r0 * aw, -CTR_CLAMP), CTR_CLAMP);
        float oy = fminf(fmaxf(r1 * ah, -CTR_CLAMP), CTR_CLAMP);
        float ctrx = cx + ox, ctry = cy + oy;
        float w = aw * expf(fminf(r2, SCALE_CLAMP));
        float h = ah * expf(fminf(r3, SCALE_CLAMP));
        float x1 = ctrx - 0.5f * w, y1 = ctry - 0.5f * h;
        float x2 = ctrx + 0.5f * w, y2 = ctry + 0.5f * h;
        prob_ws[tid]  = s;
        label_ws[tid] = label;
        boxes_ws[tid * 4 + 0] = x1; boxes_ws[tid * 4 + 1] = y1;
        boxes_ws[tid * 4 + 2] = x2; boxes_ws[tid * 4 + 3] = y2;
        float off = (float)label * CLASS_OFFSET;
        float ox1 = x1 + off, oy1 = y1 + off, ox2 = x2 + off, oy2 = y2 + off;
        boxoff_ws[tid * 4 + 0] = ox1; boxoff_ws[tid * 4 + 1] = oy1;
        boxoff_ws[tid * 4 + 2] = ox2; boxoff_ws[tid * 4 + 3] = oy2;
        areas_ws[tid] = (ox2 - ox1) * (oy2 - oy1);
    } else {
        prob_ws[tid] = 0.0f;
        label_ws[tid] = 0;
        boxes_ws[tid * 4 + 0] = 0.0f; boxes_ws[tid * 4 + 1] = 0.0f;
        boxes_ws[tid * 4 + 2] = 0.0f; boxes_ws[tid * 4 + 3] = 0.0f;
        boxoff_ws[tid * 4 + 0] = 0.0f; boxoff_ws[tid * 4 + 1] = 0.0f;
        boxoff_ws[tid * 4 + 2] = 0.0f; boxoff_ws[tid * 4 + 3] = 0.0f;
        areas_ws[tid] = 0.0f;
    }
}

// ---------------- Stage 5: IoU tiles via WMMA + async-to-LDS staging ----------------
// D = A x B with A rows = [area_i, 1, 0, 0], B rows = [1; area_j; 0; 0]
// => D[i][j] = area_i + area_j  (exact: fma(a,1,b)=a+b, other terms are 0)
// iou > t  <=>  inter > t*(sum - inter + eps)   (denominator strictly positive)
__global__ __launch_bounds__(32) void kIouWmma(
    const float4* __restrict__ boxoff, const float* __restrict__ areas,
    unsigned* __restrict__ supmask)
{
    __shared__ float4 sI[16];   // i-tile boxes
    __shared__ float4 sJ[32];   // j-tile boxes (both subtiles)

    const int lane = threadIdx.x;
    const int i0 = blockIdx.x * 16;   // row tile (16 rows)
    const int jt = blockIdx.y;        // word tile (32 cols)
    const int j0 = jt * 32;

    // Async-copy box tiles into LDS (CDNA5 ASYNCcnt path), one B128 per lane.
    // Flat LDS pointers truncate to addr[31:0] == LDS byte address (ISA 10.2).
    {
        unsigned ldsJ = (unsigned)(size_t)(&sJ[lane]);
        const float4* gJ = boxoff + (j0 + lane);
        asm volatile("global_load_async_to_lds_b128 %0, %1, off"
                     :: "v"(ldsJ), "v"(gJ) : "memory");
        if (lane < 16) {
            unsigned ldsI = (unsigned)(size_t)(&sI[lane]);
            const float4* gI = boxoff + (i0 + lane);
            asm volatile("global_load_async_to_lds_b128 %0, %1, off"
                         :: "v"(ldsI), "v"(gI) : "memory");
        }
        asm volatile("s_wait_asynccnt 0" ::: "memory");
    }
    __syncthreads();

    // A operand: 16x4 f32, lanes 0-15 hold K=0 (v0) / K=1 (v1); lanes 16-31 hold K=2/3 (zeros)
    v2f A;
    if (lane < 16) { A[0] = areas[i0 + lane]; A[1] = 1.0f; }
    else           { A[0] = 0.0f;             A[1] = 0.0f; }

    unsigned bal0[8], bal1[8];

    #pragma unroll
    for (int sub = 0; sub < 2; ++sub) {
        const int jb = j0 + sub * 16;
        // B operand: 4x16 f32, row0 = ones, row1 = area_j, rows 2-3 zero
        v2f B;
        if (lane < 16) { B[0] = 1.0f; B[1] = areas[jb + lane]; }
        else           { B[0] = 0.0f; B[1] = 0.0f; }
        v8f C = {};
        v8f D = __builtin_amdgcn_wmma_f32_16x16x4_f32(
            /*neg_a=*/false, A, /*neg_b=*/false, B,
            /*c_mod=*/(short)0, C, /*reuse_a=*/false, /*reuse_b=*/false);

        const int j = jb + (lane & 15);
        const float4 bj = sJ[sub * 16 + (lane & 15)];
        #pragma unroll
        for (int r = 0; r < 8; ++r) {
            const int ii = i0 + r + ((lane >= 16) ? 8 : 0);
            const float4 bi = sI[r + ((lane >= 16) ? 8 : 0)];
            float xx1 = fmaxf(bi.x, bj.x), yy1 = fmaxf(bi.y, bj.y);
            float xx2 = fminf(bi.z, bj.z), yy2 = fminf(bi.w, bj.w);
            float w = fmaxf(1e-28f, xx2 - xx1);
            float h = fmaxf(1e-28f, yy2 - yy1);
            float inter = w * h;
            float denom = D[r] - inter + 1e-14f;            // area_i + area_j - inter + eps > 0
            bool pred = (inter > NMS_THRESH * denom) && (j > ii) && (j < TOPK);
            unsigned b = __builtin_amdgcn_ballot_w32(pred);
            if (sub == 0) bal0[r] = b; else bal1[r] = b;
        }
    }

    if (lane == 0) {
        #pragma unroll
        for (int r = 0; r < 8; ++r) {
            // ballot bits: lanes 0-15 -> row r cols 0-15 ; lanes 16-31 -> row 8+r cols 0-15
            unsigned w0 = (bal0[r] & 0xFFFFu) | ((bal1[r] & 0xFFFFu) << 16);
            unsigned w1 = (bal0[r] >> 16)     | (bal1[r] & 0xFFFF0000u);
            supmask[(i0 + r) * 32 + jt]     = w0;
            supmask[(i0 + 8 + r) * 32 + jt] = w1;
        }
    }
}

// ---------------- Stage 6: greedy NMS over the bitmask (one wave) ----------------
__global__ __launch_bounds__(32) void kNms(const float* __restrict__ prob_ws,
                                           const unsigned* __restrict__ supmask,
                                           unsigned* __restrict__ keepw)
{
    const int t = threadIdx.x;   // lane t owns keep bits [t*32, t*32+31]
    unsigned kw = 0;
    #pragma unroll 1
    for (int b = 0; b < 32; ++b) {
        int i = t * 32 + b;
        if (i < TOPK && prob_ws[i] > CONF_THRESH) kw |= (1u << b);
    }
    for (int i = 0; i < TOPK; ++i) {
        if (i + 1 < TOPK)
            __builtin_prefetch(&supmask[(i + 1) * 32 + t], 0, 3);  // global_prefetch_b8
        unsigned w = __shfl(kw, i >> 5, 32);   // broadcast pivot word (uniform)
        if ((w >> (i & 31)) & 1u) {
            kw &= ~supmask[i * 32 + t];        // sup row already restricted to j>i
        }
    }
    keepw[t] = kw;
}

// ---------------- Stage 7: final masked outputs ----------------
__global__ void kFinal(const float* __restrict__ boxes_ws, const float* __restrict__ prob_ws,
                       const int* __restrict__ label_ws, const unsigned* __restrict__ keepw,
                       const int* __restrict__ img_h_p, const int* __restrict__ img_w_p,
                       float* __restrict__ out)
{
    int i = blockIdx.x * blockDim.x + threadIdx.x;
    if (i >= TOPK) return;
    float kf = ((keepw[i >> 5] >> (i & 31)) & 1u) ? 1.0f : 0.0f;
    float iw = (float)(*img_w_p);
    float ih = (float)(*img_h_p);
    float b0 = fminf(fmaxf(boxes_ws[i * 4 + 0] / iw, 0.0f), 1.0f);
    float b1 = fminf(fmaxf(boxes_ws[i * 4 + 1] / ih, 0.0f), 1.0f);
    float b2 = fminf(fmaxf(boxes_ws[i * 4 + 2] / iw, 0.0f), 1.0f);
    float b3 = fminf(fmaxf(boxes_ws[i * 4 + 3] / ih, 0.0f), 1.0f);
    out[i * 4 + 0] = b0 * kf;
    out[i * 4 + 1] = b1 * kf;
    out[i * 4 + 2] = b2 * kf;
    out[i * 4 + 3] = b3 * kf;
    out[TOPK * 4 + i]            = prob_ws[i] * kf;     // prob
    out[TOPK * 4 + TOPK + i]     = (float)label_ws[i];  // labels
    out[TOPK * 4 + 2 * TOPK + i] = kf;                  // keep
}

extern "C" void kernel_launch(void* const* d_in, const int* in_sizes, int n_in,
                              void* d_out, int out_size, void* d_ws, size_t ws_size,
                              hipStream_t stream) {
    const float* cls_pred    = (const float*)d_in[0];
    const float* reg_pred    = (const float*)d_in[1];
    const float* anchor_size = (const float*)d_in[2];
    const int*   fmp_w_p     = (const int*)d_in[4];
    const int*   img_h_p     = (const int*)d_in[5];
    const int*   img_w_p     = (const int*)d_in[6];
    float* out = (float*)d_out;

    const int nScores = in_sizes[0];      // m * 80
    const int ka = in_sizes[2] / 2;       // anchors per cell

    // workspace layout (bytes)
    char* ws = (char*)d_ws;
    int*                hist      = (int*)(ws + 0);                 // 4096 ints
    int*                counter   = (int*)(ws + 16384);
    int*                cutoff    = (int*)(ws + 16388);
    unsigned long long* cand      = (unsigned long long*)(ws + 16416); // 4096 u64
    float*              prob_ws   = (float*)(ws + 49184);           // 1024
    int*                label_ws  = (int*)(ws + 53280);             // 1024
    float*              boxes_ws  = (float*)(ws + 57376);           // 1024*4
    float*              boxoff_ws = (float*)(ws + 73760);           // 1024*4 (16B aligned)
    float*              areas_ws  = (float*)(ws + 90144);           // 1024
    unsigned*           supmask   = (unsigned*)(ws + 94240);        // 1008*32 words
    unsigned*           keepw     = (unsigned*)(ws + 94240 + 1008 * 32 * 4); // 32 words

    kZero<<<(NBINS + 1 + 255) / 256, 256, 0, stream>>>(hist, counter);

    int blocksN = (nScores + 255) / 256;
    kHist<<<blocksN, 256, 0, stream>>>(cls_pred, nScores, hist);
    kCutoff<<<1, 1, 0, stream>>>(hist, cutoff);
    kCollect<<<blocksN, 256, 0, stream>>>(cls_pred, nScores, cutoff, cand, counter);

    kSortDecode<<<1, 1024, 0, stream>>>(cand, counter, reg_pred, anchor_size, fmp_w_p, ka,
                                        prob_ws, label_ws, boxes_ws, boxoff_ws, areas_ws);

    dim3 iouGrid((TOPK + 15) / 16, (TOPK + 31) / 32);  // 63 x 32
    kIouWmma<<<iouGrid, 32, 0, stream>>>((const float4*)boxoff_ws, areas_ws, supmask);

    kNms<<<1, 32, 0, stream>>>(prob_ws, supmask, keepw);

    kFinal<<<(TOPK + 255) / 256, 256, 0, stream>>>(boxes_ws, prob_ws, label_ws, keepw,
                                                   img_h_p, img_w_p, out);
}